// ZERON_GCN_edge_70609262346361
// MI455X (gfx1250) — compile-verified
//
#include <hip/hip_runtime.h>

// Problem constants (match reference setup_inputs)
#define NNODES 100000
#define NEDGES 1600000
#define DIN    128
#define DOUT   128
#define D10    12          // DOUT // 10

typedef float v2f __attribute__((ext_vector_type(2)));
typedef float v8f __attribute__((ext_vector_type(8)));

// ---------------------------------------------------------------------------
// Kernel 1: support = x @ W  via native fp32 WMMA (v_wmma_f32_16x16x4_f32).
// One wave -> one 16x16 output tile. Block = 256 threads = 8 waves = all 8
// column tiles of one 16-row stripe. grid.x = N/16 = 6250 (exact).
//
// fp32 A 16x4 layout (ISA 7.12.2): lane L<16 holds A[M=L][K=0..1] in v0,v1;
// lane L>=16 holds A[M=L-16][K=2..3]. B 4x16 mirrors with N = lane&15.
// C/D: VGPR r -> M = r + 8*(lane>>4), N = lane&15.
// ---------------------------------------------------------------------------
__global__ void __launch_bounds__(256)
gcn_gemm_wmma_f32(const float* __restrict__ x,
                  const float* __restrict__ W,
                  float* __restrict__ support /* == d_out */) {
    const int lane = threadIdx.x & 31;
    const int wave = threadIdx.x >> 5;      // 0..7 -> column tile
    const int half = lane >> 4;             // 0 or 1
    const int l16  = lane & 15;
    const int row0 = blockIdx.x << 4;       // 16-row stripe
    const int col0 = wave << 4;

    // per-lane base pointers
    const float* xrow = x + (size_t)(row0 + l16) * DIN + 2 * half;      // A[M=l16][k+2*half]
    const float* wcol = W + (size_t)(2 * half) * DOUT + col0 + l16;     // B[k+2*half][N=l16]

    v8f c = {0.f, 0.f, 0.f, 0.f, 0.f, 0.f, 0.f, 0.f};

#pragma unroll
    for (int k = 0; k < DIN; k += 4) {
        v2f a, b;
        a.x = xrow[k];                  // x[row][k + 2*half]     (contiguous pair -> b64 load)
        a.y = xrow[k + 1];              // x[row][k + 2*half + 1]
        b.x = wcol[(size_t)k * DOUT];         // W[k + 2*half][col]
        b.y = wcol[(size_t)(k + 1) * DOUT];   // W[k + 2*half + 1][col]
        c = __builtin_amdgcn_wmma_f32_16x16x4_f32(
                /*neg_a=*/false, a, /*neg_b=*/false, b,
                /*c_mod=*/(short)0, c, /*reuse_a=*/false, /*reuse_b=*/false);
    }

    float* o = support + (size_t)row0 * DOUT + col0 + l16;
#pragma unroll
    for (int r = 0; r < 8; ++r)
        o[(size_t)(r + 8 * half) * DOUT] = c[r];
}

// ---------------------------------------------------------------------------
// Kernel 2: zero workspace (agg12: N*D10 floats, deg: N floats) each call.
// ---------------------------------------------------------------------------
__global__ void gcn_zero_ws(float* __restrict__ ws, int count) {
    int t = blockIdx.x * blockDim.x + threadIdx.x;
    if (t < count) ws[t] = 0.0f;
}

// ---------------------------------------------------------------------------
// Kernel 3: edge aggregation — only the first D10=12 columns are ever used
// downstream, so gather/scatter 12 floats per edge instead of 128 (10x less
// HBM traffic on the dominant phase). 16 lanes per edge: lanes 0..11 move
// columns (coalesced 48B gather + 48B atomic scatter), lane 12 counts degree.
// ---------------------------------------------------------------------------
__global__ void __launch_bounds__(256)
gcn_edge_agg(const float* __restrict__ support,   // d_out (written by GEMM)
             const int* __restrict__ row,
             const int* __restrict__ col,
             float* __restrict__ agg12,           // [N, D10]
             float* __restrict__ deg) {           // [N]
    int t = blockIdx.x * blockDim.x + threadIdx.x;
    int e = t >> 4;
    int c = t & 15;
    if (e >= NEDGES) return;
    int r = row[e];
    if (c < D10) {
        int s = col[e];
        float v = support[(size_t)s * DOUT + c];
        atomicAdd(agg12 + (size_t)r * D10 + c, v);
    } else if (c == D10) {
        atomicAdd(deg + r, 1.0f);
    }
}

// ---------------------------------------------------------------------------
// Kernel 4: finalize in place over d_out:
//   h[:, :12]  = agg12/deg + bias[:12]
//   h[:, 12:]  = support + bias[12:]
//   out        = PReLU(h, alpha)      (per-channel alpha)
// ---------------------------------------------------------------------------
__global__ void __launch_bounds__(256)
gcn_finalize(float* __restrict__ out,             // d_out: support in, result out
             const float* __restrict__ agg12,
             const float* __restrict__ deg,
             const float* __restrict__ bias,
             const float* __restrict__ alpha) {
    int t = blockIdx.x * blockDim.x + threadIdx.x;
    if (t >= NNODES * DOUT) return;
    int n = t >> 7;         // DOUT == 128
    int c = t & 127;
    float v;
    if (c < D10) {
        v = agg12[(size_t)n * D10 + c] / deg[n];
    } else {
        v = out[t];
    }
    float h = v + bias[c];
    out[t] = (h >= 0.0f) ? h : alpha[c] * h;
}

// ---------------------------------------------------------------------------
extern "C" void kernel_launch(void* const* d_in, const int* in_sizes, int n_in,
                              void* d_out, int out_size, void* d_ws, size_t ws_size,
                              hipStream_t stream) {
    const float* x     = (const float*)d_in[0];   // [N, DIN]
    const float* W     = (const float*)d_in[1];   // [DIN, DOUT]
    const float* bias  = (const float*)d_in[2];   // [DOUT]
    const float* alpha = (const float*)d_in[3];   // [DOUT]
    const int*   eidx  = (const int*)d_in[4];     // [2, E] -> row = eidx, col = eidx + E
    (void)in_sizes; (void)n_in; (void)out_size; (void)ws_size;

    const int* row = eidx;
    const int* col = eidx + NEDGES;

    float* support = (float*)d_out;               // [N, DOUT], reused as output
    float* agg12   = (float*)d_ws;                // [N, D10]
    float* deg     = agg12 + (size_t)NNODES * D10;// [N]

    // 1) zero accumulators (every call; capture-safe)
    {
        int cnt = NNODES * (D10 + 1);
        gcn_zero_ws<<<(cnt + 255) / 256, 256, 0, stream>>>((float*)d_ws, cnt);
    }

    // 2) dense GEMM via fp32 WMMA: 6250 blocks x 8 waves, each wave a 16x16 tile
    gcn_gemm_wmma_f32<<<NNODES / 16, 256, 0, stream>>>(x, W, support);

    // 3) edge aggregation over first 12 columns only
    {
        long long threads = (long long)NEDGES * 16;
        int blocks = (int)((threads + 255) / 256);
        gcn_edge_agg<<<blocks, 256, 0, stream>>>(support, row, col, agg12, deg);
    }

    // 4) divide / concat / bias / PReLU, in place on d_out
    {
        int cnt = NNODES * DOUT;
        gcn_finalize<<<(cnt + 255) / 256, 256, 0, stream>>>(support, agg12, deg, bias, alpha);
    }
}